// PyramidROIAlign_48627619725488
// MI455X (gfx1250) — compile-verified
//
#include <hip/hip_runtime.h>
#include <hip/hip_bf16.h>
#include <cmath>

// PyramidROIAlign for MI455X (gfx1250, wave32).
// Bilinear crop-and-resize expressed as a block-sparse GEMM evaluated with
// V_WMMA_F32_16X16X4_F32:  out_row[7 x 256] = W[7 x 28] * G[28 x 256]
// where G = gathered corner samples (2 y-taps x 14 x-taps) for this output row.
// Memory-bound (~500 MB @ 23.3 TB/s => ~21us floor); WMMA does the lerp math
// while loads stay channel-coalesced. j/tile loops kept rolled to hold VGPR
// count < 128 so each SIMD32 carries enough waves to hide gather latency
// (the fully-unrolled version spilled past v256 and paid s_set_vgpr_msb tax).

typedef __attribute__((ext_vector_type(2))) float v2f;
typedef __attribute__((ext_vector_type(8))) float v8f;

#define POOLN 7
#define NTAPS 28          // per output row: 2 y-taps * (7 px * 2 x-taps)
#define CCH   256

__global__ __launch_bounds__(256)
void roialign_wmma_kernel(const float* __restrict__ boxes,
                          const float* __restrict__ image_shape,
                          const float* __restrict__ P2,
                          const float* __restrict__ P3,
                          const float* __restrict__ P4,
                          const float* __restrict__ P5,
                          float* __restrict__ out,
                          int nroi_per_batch)
{
    __shared__ int   s_goff[POOLN][NTAPS];  // element offset (y*W + x)*C into level fmap
    __shared__ float s_wgt [POOLN][NTAPS];  // bilinear weight for that tap

    const int roi = blockIdx.x;
    const int b   = roi / nroi_per_batch;

    // ---- ROI level selection (uniform across block) ----
    const float by0 = boxes[roi * 4 + 0];
    const float bx0 = boxes[roi * 4 + 1];
    const float by1 = boxes[roi * 4 + 2];
    const float bx1 = boxes[roi * 4 + 3];

    const float area = image_shape[0] * image_shape[1];
    const float hh = by1 - by0;
    const float wwb = bx1 - bx0;
    const float lv = log2f(sqrtf(fmaxf(hh * wwb, 1e-20f)) * (sqrtf(area) * (1.0f / 224.0f)));
    int lvl = 4 + (int)rintf(lv);
    lvl = lvl < 2 ? 2 : (lvl > 5 ? 5 : lvl);

    const float* fmap;
    int Hd;
    switch (lvl) {
        case 2:  fmap = P2; Hd = 256; break;
        case 3:  fmap = P3; Hd = 128; break;
        case 4:  fmap = P4; Hd = 64;  break;
        default: fmap = P5; Hd = 32;  break;
    }
    const int Wd = Hd;
    fmap += (size_t)b * (size_t)Hd * (size_t)Wd * CCH;

    // ---- setup: one thread per (row j, tap r) builds offset + weight tables ----
    {
        const int t = threadIdx.x;
        if (t < POOLN * NTAPS) {
            const int j    = t / NTAPS;
            const int r    = t % NTAPS;
            const int ysel = r / 14;        // 0 = top tap, 1 = bottom tap
            const int xsel = r % 14;
            const int px   = xsel >> 1;     // which output pixel this tap feeds
            const int odd  = xsel & 1;      // 0 = left tap, 1 = right tap

            const float fy = (float)j  * (1.0f / 6.0f);
            const float fx = (float)px * (1.0f / 6.0f);

            const float ys = (by0 + fy * (by1 - by0)) * (float)(Hd - 1);
            const float xs = (bx0 + fx * (bx1 - bx0)) * (float)(Wd - 1);

            const float yf = floorf(ys);
            const float xf = floorf(xs);
            const float wy = ys - yf;
            const float wx = xs - xf;

            int y0 = (int)yf;  y0 = y0 < 0 ? 0 : (y0 > Hd - 1 ? Hd - 1 : y0);
            int x0 = (int)xf;  x0 = x0 < 0 ? 0 : (x0 > Wd - 1 ? Wd - 1 : x0);
            const int y1i = y0 + 1 > Hd - 1 ? Hd - 1 : y0 + 1;
            const int x1i = x0 + 1 > Wd - 1 ? Wd - 1 : x0 + 1;

            const int yy = ysel ? y1i : y0;
            const int xx = odd  ? x1i : x0;

            s_goff[j][r] = (yy * Wd + xx) * CCH;
            s_wgt [j][r] = (ysel ? wy : 1.0f - wy) * (odd ? wx : 1.0f - wx);
        }
    }
    __syncthreads();

    // ---- main: 8 waves, wave w owns channels [32w, 32w+32) as two N=16 tiles ----
    const int lane   = threadIdx.x & 31;
    const int wv     = threadIdx.x >> 5;   // 0..7
    const int laneLo = lane & 15;
    const int hi     = lane >> 4;          // 0: K 0/1 half, 1: K 2/3 half
    const int p      = laneLo;             // A-matrix M row (output pixel; >=7 -> padding)

#pragma unroll 1
    for (int tile = 0; tile < 2; ++tile) {
        const int c = wv * 32 + tile * 16 + laneLo;   // channel (N column)

#pragma unroll 1
        for (int j = 0; j < POOLN; ++j) {
            v8f acc = {0.f, 0.f, 0.f, 0.f, 0.f, 0.f, 0.f, 0.f};

            // prefetch next row's first taps for this lane's channel
            if (j < POOLN - 1) {
                __builtin_prefetch(fmap + s_goff[j + 1][2 * hi] + c, 0, 1);
            }

#pragma unroll
            for (int k = 0; k < 7; ++k) {             // K = 28 taps, 4 per WMMA
                const int r0 = 4 * k + 2 * hi;        // this lane's two K rows
                const int r1 = r0 + 1;

                // A: weight matrix slice (nonzero only when tap feeds pixel p)
                const float w0 = s_wgt[j][r0];
                const float w1 = s_wgt[j][r1];
                const int   i0 = (r0 % 14) >> 1;
                const int   i1 = (r1 % 14) >> 1;
                v2f a;
                a.x = (i0 == p) ? w0 : 0.0f;
                a.y = (i1 == p) ? w1 : 0.0f;

                // B: gathered samples, channel-coalesced across lanes
                v2f bv;
                bv.x = fmap[s_goff[j][r0] + c];
                bv.y = fmap[s_goff[j][r1] + c];

                acc = __builtin_amdgcn_wmma_f32_16x16x4_f32(
                          /*neg_a=*/false, a, /*neg_b=*/false, bv,
                          /*c_mod=*/(short)0, acc,
                          /*reuse_a=*/false, /*reuse_b=*/false);
            }

            // D layout: lanes 0-15 hold px = v (v=0..6 valid); lanes 16-31 hold px = 8+v (padding)
            if (hi == 0) {
                float* op = out + ((size_t)roi * 49 + (size_t)j * 7) * CCH
                                + (size_t)wv * 32 + (size_t)tile * 16 + laneLo;
#pragma unroll
                for (int v = 0; v < 7; ++v) {
                    op[v * CCH] = acc[v];
                }
            }
        }
    }
}

extern "C" void kernel_launch(void* const* d_in, const int* in_sizes, int n_in,
                              void* d_out, int out_size, void* d_ws, size_t ws_size,
                              hipStream_t stream)
{
    const float* boxes = (const float*)d_in[0];
    const float* ishp  = (const float*)d_in[1];
    const float* P2    = (const float*)d_in[2];
    const float* P3    = (const float*)d_in[3];
    const float* P4    = (const float*)d_in[4];
    const float* P5    = (const float*)d_in[5];
    float* out         = (float*)d_out;

    // B from P2 size; N from boxes size.
    const int B = in_sizes[2] / (256 * 256 * 256);
    const int total_rois = in_sizes[0] / 4;        // B * N
    const int nroi_per_batch = total_rois / (B > 0 ? B : 1);

    roialign_wmma_kernel<<<total_rois, 256, 0, stream>>>(
        boxes, ishp, P2, P3, P4, P5, out, nroi_per_batch);
}